// ScaledDotProductAttention_66907000537739
// MI455X (gfx1250) — compile-verified
//
#include <hip/hip_runtime.h>
#include <math.h>

// Problem dims (fixed by the reference)
#define KB   2      // batch
#define AQ   2048   // q_len
#define HH   16     // heads
#define DK   64     // head dim
#define R1   64
#define R2   16
#define MM   2048   // kv_len
#define DV   64

typedef __attribute__((ext_vector_type(2))) float v2f;
typedef __attribute__((ext_vector_type(8))) float v8f;

// ---------------------------------------------------------------------------
// Kernel 1: collapse the low-rank cascade into one 64x64 matrix per head:
//   Wc[h] = W_A[h] @ W_B[h] @ W_Bt[h] @ W_At[h]
// Tiny (16 blocks x 64 threads); LDS staging for the intermediates.
// ---------------------------------------------------------------------------
__global__ void wcomb_kernel(const float* __restrict__ W_A,
                             const float* __restrict__ W_B,
                             const float* __restrict__ W_Bt,
                             const float* __restrict__ W_At,
                             float* __restrict__ Wc) {
    const int h = blockIdx.x;
    const int t = threadIdx.x;            // 0..63, one output row each
    __shared__ float T1[DK * R2];         // W_A @ W_B   : [64,16]
    __shared__ float T2[DK * R1];         // T1 @ W_Bt   : [64,64]

    const float* wa  = W_A  + (size_t)h * DK * R1;   // [DK,R1]
    const float* wb  = W_B  + (size_t)h * R1 * R2;   // [R1,R2]
    const float* wbt = W_Bt + (size_t)h * R2 * R1;   // [R2,R1]
    const float* wat = W_At + (size_t)h * R1 * DK;   // [R1,DK]

    #pragma unroll 4
    for (int j = 0; j < R2; ++j) {
        float s = 0.f;
        for (int c = 0; c < R1; ++c) s += wa[t * R1 + c] * wb[c * R2 + j];
        T1[t * R2 + j] = s;
    }
    __syncthreads();
    #pragma unroll 4
    for (int j = 0; j < R1; ++j) {
        float s = 0.f;
        for (int c = 0; c < R2; ++c) s += T1[t * R2 + c] * wbt[c * R1 + j];
        T2[t * R1 + j] = s;
    }
    __syncthreads();
    #pragma unroll 4
    for (int j = 0; j < DK; ++j) {
        float s = 0.f;
        for (int c = 0; c < R1; ++c) s += T2[t * R1 + c] * wat[c * DK + j];
        Wc[(size_t)h * DK * DK + t * DK + j] = s;
    }
}

// ---------------------------------------------------------------------------
// Kernel 2: qproj[k,h,a,:] = q[k,a,h,:] @ Wc[h]   (also transposes h<->a so
// stage-3 reads contiguous [K,H,A,DK] rows). Wc[h] staged in LDS (16 KB).
// ---------------------------------------------------------------------------
__global__ void qproj_kernel(const float* __restrict__ q,
                             const float* __restrict__ Wc,
                             float* __restrict__ qproj) {
    const int kh = blockIdx.y;
    const int k  = kh / HH, h = kh % HH;
    const int a  = blockIdx.x * 128 + threadIdx.x;

    __shared__ float wc[DK * DK];
    for (int i = threadIdx.x; i < DK * DK; i += 128)
        wc[i] = Wc[(size_t)h * DK * DK + i];
    __syncthreads();

    const float* qr = q + (((size_t)k * AQ + a) * HH + h) * DK;
    float r[DK];
    #pragma unroll
    for (int i = 0; i < DK / 4; ++i) {
        float4 f = ((const float4*)qr)[i];
        r[4*i+0] = f.x; r[4*i+1] = f.y; r[4*i+2] = f.z; r[4*i+3] = f.w;
    }
    float* o = qproj + (((size_t)k * HH + h) * AQ + a) * DK;
    #pragma unroll 4
    for (int j = 0; j < DK; ++j) {
        float s = 0.f;
        #pragma unroll
        for (int d = 0; d < DK; ++d) s += r[d] * wc[d * DK + j];  // broadcast
        o[j] = s;
    }
}

// ---------------------------------------------------------------------------
// Kernel 3: raw scores = qproj @ qt per (k,h), via V_WMMA_F32_16X16X4_F32.
// Block = 256 threads (8 waves); each wave owns a 16-row A tile and sweeps
// 16-wide M chunks. Stores raw scores into the attn output region.
//   A frag (16x4 f32): lanes0-15 K={0,1}, lanes16-31 K={2,3}  (v2f)
//   B frag (4x16 f32): VGPR0 K = lane<16?0:2, VGPR1 K = lane<16?1:3
//   C/D   (16x16 f32): VGPR r -> row r + 8*(lane>=16), col lane&15
// ---------------------------------------------------------------------------
__global__ void scores_kernel(const float* __restrict__ qproj, // [K,H,A,DK]
                              const float* __restrict__ qt,    // [K,H,DK,M]
                              float* __restrict__ attn) {      // [K,H,A,M]
    const int kh    = blockIdx.y;
    const int arow0 = blockIdx.x * 16;
    const int wave  = threadIdx.x >> 5;
    const int lane  = threadIdx.x & 31;
    const int half  = lane >> 4;
    const int l16   = lane & 15;

    const float* Qp  = qproj + ((size_t)kh * AQ + arow0) * DK;
    const float* Qt  = qt    + (size_t)kh * DK * MM;
    float*       Out = attn  + ((size_t)kh * AQ + arow0) * MM;

    // Load the 16x64 A tile once: 16 K-steps of v2f.
    v2f afrag[16];
    #pragma unroll
    for (int s = 0; s < 16; ++s) {
        const int kb = 4 * s + 2 * half;
        afrag[s].x = Qp[l16 * DK + kb + 0];
        afrag[s].y = Qp[l16 * DK + kb + 1];
    }

    for (int mc = wave; mc < MM / 16; mc += 8) {   // uniform per wave
        const int m0 = mc * 16;
        v8f c = {};
        #pragma unroll
        for (int s = 0; s < 16; ++s) {
            const int kb = 4 * s + 2 * half;
            v2f b;
            b.x = Qt[(size_t)(kb + 0) * MM + m0 + l16];
            b.y = Qt[(size_t)(kb + 1) * MM + m0 + l16];
            c = __builtin_amdgcn_wmma_f32_16x16x4_f32(
                    false, afrag[s], false, b, (short)0, c, false, false);
        }
        #pragma unroll
        for (int r = 0; r < 8; ++r) {
            const int row = r + 8 * half;
            Out[(size_t)row * MM + m0 + l16] = c[r];
        }
    }
}

// ---------------------------------------------------------------------------
// Kernel 4: in-place row softmax over M=2048. One wave per row (64 f32/lane
// held in registers), cross-lane max/sum via wave32 xor-shuffles.
// ---------------------------------------------------------------------------
__global__ void softmax_kernel(float* __restrict__ attn) {
    const size_t row  = (size_t)blockIdx.x * 8 + (threadIdx.x >> 5);
    const int    lane = threadIdx.x & 31;
    float* p = attn + row * MM;

    float vals[64];
    float mx = -INFINITY;
    #pragma unroll
    for (int i = 0; i < 16; ++i) {
        float4 f = ((const float4*)p)[lane + 32 * i];
        vals[4*i+0] = f.x; vals[4*i+1] = f.y; vals[4*i+2] = f.z; vals[4*i+3] = f.w;
        mx = fmaxf(mx, fmaxf(fmaxf(f.x, f.y), fmaxf(f.z, f.w)));
    }
    #pragma unroll
    for (int off = 16; off > 0; off >>= 1)
        mx = fmaxf(mx, __shfl_xor(mx, off, 32));

    float sum = 0.f;
    #pragma unroll
    for (int i = 0; i < 64; ++i) {
        vals[i] = __expf(vals[i] - mx);
        sum += vals[i];
    }
    #pragma unroll
    for (int off = 16; off > 0; off >>= 1)
        sum += __shfl_xor(sum, off, 32);

    const float inv = 1.f / sum;
    #pragma unroll
    for (int i = 0; i < 16; ++i) {
        float4 f;
        f.x = vals[4*i+0] * inv; f.y = vals[4*i+1] * inv;
        f.z = vals[4*i+2] * inv; f.w = vals[4*i+3] * inv;
        ((float4*)p)[lane + 32 * i] = f;
    }
}

// ---------------------------------------------------------------------------
// Kernel 5: out = attn @ v per (k,h), WMMA f32 16x16x4, K-loop over M=2048.
// One wave per 16-row tile; 4 accumulators cover DV=64. v (17 MB total) is
// L2-resident so the 128x per-(k,h) re-read stays on-chip.
// ---------------------------------------------------------------------------
__global__ void pv_kernel(const float* __restrict__ attn, // [K,H,A,M]
                          const float* __restrict__ v,    // [K,H,M,DV]
                          float* __restrict__ out) {      // [K,H,A,DV]
    const int kh    = blockIdx.y;
    const int arow0 = blockIdx.x * 16;
    const int lane  = threadIdx.x & 31;
    const int half  = lane >> 4;
    const int l16   = lane & 15;

    const float* P = attn + ((size_t)kh * AQ + arow0) * MM;
    const float* V = v    + (size_t)kh * MM * DV;
    float*       O = out  + ((size_t)kh * AQ + arow0) * DV;

    v8f acc[4] = {v8f{}, v8f{}, v8f{}, v8f{}};

    #pragma unroll 4
    for (int mb = 0; mb < MM; mb += 4) {
        const int kb = mb + 2 * half;
        v2f a;
        a.x = P[(size_t)l16 * MM + kb + 0];
        a.y = P[(size_t)l16 * MM + kb + 1];
        #pragma unroll
        for (int n = 0; n < 4; ++n) {
            v2f b;
            b.x = V[(size_t)(kb + 0) * DV + n * 16 + l16];
            b.y = V[(size_t)(kb + 1) * DV + n * 16 + l16];
            acc[n] = __builtin_amdgcn_wmma_f32_16x16x4_f32(
                         false, a, false, b, (short)0, acc[n], false, false);
        }
    }
    #pragma unroll
    for (int n = 0; n < 4; ++n) {
        #pragma unroll
        for (int r = 0; r < 8; ++r) {
            const int row = r + 8 * half;
            O[(size_t)row * DV + n * 16 + l16] = acc[n][r];
        }
    }
}

// ---------------------------------------------------------------------------
extern "C" void kernel_launch(void* const* d_in, const int* in_sizes, int n_in,
                              void* d_out, int out_size, void* d_ws, size_t ws_size,
                              hipStream_t stream) {
    const float* q    = (const float*)d_in[0];  // [K,A,H,DK]
    const float* W_A  = (const float*)d_in[1];  // [H,DK,R1]
    const float* W_B  = (const float*)d_in[2];  // [H,R1,R2]
    const float* W_Bt = (const float*)d_in[3];  // [H,R2,R1]
    const float* W_At = (const float*)d_in[4];  // [H,R1,DK]
    const float* qt   = (const float*)d_in[5];  // [K,H,DK,M]
    const float* v    = (const float*)d_in[6];  // [K,H,M,DV]
    // d_in[7] = d_k (scalar; reference never uses it for scaling)

    float* out  = (float*)d_out;                          // [K,H,A,DV]
    float* attn = out + (size_t)KB * HH * AQ * DV;        // [K,H,A,M]

    // workspace: Wc (256 KB) + qproj (16.8 MB)
    float* Wc    = (float*)d_ws;
    float* qproj = Wc + (size_t)HH * DK * DK;

    // 1) collapse cascade weights per head
    wcomb_kernel<<<HH, 64, 0, stream>>>(W_A, W_B, W_Bt, W_At, Wc);

    // 2) qproj = q @ Wc, re-laid out to [K,H,A,DK]
    qproj_kernel<<<dim3(AQ / 128, KB * HH), 128, 0, stream>>>(q, Wc, qproj);

    // 3) raw scores -> attn region (WMMA)
    scores_kernel<<<dim3(AQ / 16, KB * HH), 256, 0, stream>>>(qproj, qt, attn);

    // 4) softmax in place over M
    softmax_kernel<<<(KB * HH * AQ) / 8, 256, 0, stream>>>(attn);

    // 5) out = attn @ v (WMMA)
    pv_kernel<<<dim3(AQ / 16, KB * HH), 32, 0, stream>>>(attn, v, out);
}